// CapsLayer2D_91276644974849
// MI455X (gfx1250) — compile-verified
//
#include <hip/hip_runtime.h>

// CDNA5 / gfx1250 fused capsule-routing kernel.
// Fuses u_hat GEMM (fp32 WMMA 16x16x4) + 3 dynamic-routing iterations in LDS,
// avoiding the 128MB u_hat intermediate entirely (~15.5MB total HBM traffic).

typedef __attribute__((ext_vector_type(2))) float v2f;
typedef __attribute__((ext_vector_type(8))) float v8f;

constexpr int N_IN     = 64;   // input capsules (i)
constexpr int D_IN     = 16;   // input capsule dim (e) == WMMA K
constexpr int N_CAPS   = 10;   // output capsules (n)
constexpr int CAPS_DIM = 16;   // output capsule dim (o) == WMMA N
constexpr int POS_TILE = 16;   // positions per block == WMMA M

// Pad per-position strides by 4 floats so lane-varying `pos` maps to
// different LDS banks (stride 1028 % 64 == 4) instead of all hitting bank 0.
constexpr int A_STRIDE = N_IN * D_IN + 4;      // 1028 floats / position
constexpr int R_STRIDE = N_IN * CAPS_DIM + 4;  // 1028 floats / position

constexpr size_t SMEM_FLOATS =
    (size_t)POS_TILE * A_STRIDE +    // staged input tile
    (size_t)POS_TILE * R_STRIDE +    // u_hat tile (res)
    (size_t)POS_TILE * N_IN +        // routing logits b
    (size_t)POS_TILE * CAPS_DIM;     // v scratch
constexpr size_t SMEM_BYTES = SMEM_FLOATS * sizeof(float);

__global__ __launch_bounds__(256)
void caps_routing_kernel(const float* __restrict__ inp,   // [3136][64][16]
                         const float* __restrict__ W,     // [10][64][16][16]
                         float* __restrict__ out)         // [3136][10][16]
{
    extern __shared__ float smem[];
    float* a_lds   = smem;                               // padded [16][1028]
    float* res_lds = a_lds   + POS_TILE * A_STRIDE;      // padded [16][1028]
    float* b_lds   = res_lds + POS_TILE * R_STRIDE;      // [16][64]
    float* v_lds   = b_lds   + POS_TILE * N_IN;          // [16][16]

    const int tile = blockIdx.x;       // 196 position tiles
    const int ncap = blockIdx.y;       // 10 output capsules
    const int t    = threadIdx.x;      // 256 threads = 8 wave32
    const int lane = t & 31;
    const int wave = t >> 5;
    const int m    = lane & 15;        // M (pos) / N (o) index within fragment
    const int half = lane >> 4;        // K-half selector

    const float* Wn = W + (size_t)ncap * N_IN * D_IN * CAPS_DIM;

    // Warm L2/WGP$ for this block's 64KB W[n] slice (global_prefetch_b8);
    // each wave prefetches a distinct stripe of cachelines.
    {
        const char* wp = (const char*)Wn + (size_t)t * 256;
        __builtin_prefetch(wp, 0, 3);
    }

    // ---- Stage input tile: 16 pos x 1024 floats, float4-coalesced into padded LDS
    {
        const float4* gin = (const float4*)(inp + (size_t)tile * POS_TILE * N_IN * D_IN);
        float4* sa = (float4*)a_lds;   // per-pos stride = A_STRIDE/4 = 257 float4
        #pragma unroll
        for (int k = 0; k < 16; ++k) {
            int idx = t + k * 256;     // 0..4095 (256 float4 per position)
            int pos = idx >> 8;
            int rem = idx & 255;
            sa[pos * (A_STRIDE / 4) + rem] = gin[idx];
        }
    }
    // Routing logits start uniform (softmax of zeros) = 1/64
    #pragma unroll
    for (int k = 0; k < 4; ++k) b_lds[t + k * 256] = 1.0f / (float)N_IN;
    __syncthreads();

    // ---- WMMA phase: res[pos, i, o] = sum_e A[pos, i, e] * W[n, i, e, o]
    // For each i: one 16x16x16 fp32 GEMM = 4 chained V_WMMA_F32_16X16X4_F32.
    // A frag (16x4 f32):  VGPR j -> K = kk + 2*half + j, M = m   (ISA 7.12.2)
    // B frag (4x16 f32):  VGPR j -> K = kk + 2*half + j, N = m
    // D frag (16x16 f32): VGPR r -> M = r + 8*half,      N = m
    for (int ii = 0; ii < 8; ++ii) {
        const int i = wave * 8 + ii;
        const float* arow = a_lds + m * A_STRIDE + i * D_IN;    // A[m][i][*]
        const float* wrow = Wn + i * (D_IN * CAPS_DIM) + m;     // W[n][i][*][m]
        v8f acc = {};
        #pragma unroll
        for (int kk = 0; kk < D_IN; kk += 4) {
            const int e0 = kk + half * 2;
            v2f af, bf;
            af.x = arow[e0];
            af.y = arow[e0 + 1];
            bf.x = wrow[(e0)     * CAPS_DIM];
            bf.y = wrow[(e0 + 1) * CAPS_DIM];
            acc = __builtin_amdgcn_wmma_f32_16x16x4_f32(
                /*neg_a=*/false, af, /*neg_b=*/false, bf,
                /*c_mod=*/(short)0, acc, /*reuse_a=*/false, /*reuse_b=*/false);
        }
        #pragma unroll
        for (int r = 0; r < 8; ++r)
            res_lds[(r + 8 * half) * R_STRIDE + i * CAPS_DIM + m] = acc[r];
    }
    __syncthreads();

    // ---- Routing phase: one thread per (pos, o); pos = t>>4, o = t&15.
    // Lanes [16p .. 16p+15] of a wave share one pos -> squash norm via shfl_xor.
    const int pos = t >> 4;
    const int o   = t & 15;
    const float* rrow = res_lds + pos * R_STRIDE + o;
    const float* brow = b_lds + pos * N_IN;
    float vout = 0.0f;

    for (int it = 0; it < 3; ++it) {           // N_ROUTINGS = 3
        // s[pos,o] = sum_i b[pos,i] * res[pos,i,o]
        float s = 0.0f;
        #pragma unroll 8
        for (int i = 0; i < N_IN; ++i)
            s = fmaf(brow[i], rrow[i * CAPS_DIM], s);

        // squash over o (16-lane group reduction, wave32-native)
        float sn = s * s;
        #pragma unroll
        for (int mk = 1; mk < 16; mk <<= 1)
            sn += __shfl_xor(sn, mk, 32);      // xor masks <16 stay in-group
        const float scale = sqrtf(sn) / (1.0f + sn);
        vout = scale * s;

        if (it < 2) {
            v_lds[pos * CAPS_DIM + o] = vout;
            __syncthreads();
            // agreement: b[pos,i] += sum_o res[pos,i,o] * v[pos,o]; 4 i per thread
            const int ib = (t & 15) * 4;
            const float* vrow = v_lds + pos * CAPS_DIM;
            #pragma unroll
            for (int q = 0; q < 4; ++q) {
                const float* rr = res_lds + pos * R_STRIDE + (ib + q) * CAPS_DIM;
                float dot = 0.0f;
                #pragma unroll
                for (int oo = 0; oo < CAPS_DIM; ++oo)
                    dot = fmaf(rr[oo], vrow[oo], dot);
                b_lds[pos * N_IN + ib + q] += dot;
            }
            __syncthreads();
        }
    }

    const int pos_g = tile * POS_TILE + pos;
    out[(size_t)pos_g * (N_CAPS * CAPS_DIM) + ncap * CAPS_DIM + o] = vout;
}

extern "C" void kernel_launch(void* const* d_in, const int* in_sizes, int n_in,
                              void* d_out, int out_size, void* d_ws, size_t ws_size,
                              hipStream_t stream) {
    const float* inp = (const float*)d_in[0];   // [16,14,14,64,16] fp32
    const float* W   = (const float*)d_in[1];   // [10,64,16,16]    fp32
    float* out       = (float*)d_out;           // [16,14,14,10,16] fp32

    const int npos  = in_sizes[0] / (N_IN * D_IN);   // 3136
    const int tiles = npos / POS_TILE;               // 196 (exact)

    // >64KB dynamic LDS: raise the attribute. Idempotent, not a stream op
    // (graph-capture safe), called unconditionally every launch — no static
    // guards, per harness rules.
    hipFuncSetAttribute((const void*)caps_routing_kernel,
                        hipFuncAttributeMaxDynamicSharedMemorySize,
                        (int)SMEM_BYTES);

    dim3 grid(tiles, N_CAPS);
    caps_routing_kernel<<<grid, 256, SMEM_BYTES, stream>>>(inp, W, out);
}